// YOLOLoss_84018150244766
// MI455X (gfx1250) — compile-verified
//
#include <hip/hip_runtime.h>
#include <stdint.h>

// ---------------- problem geometry ----------------
#define DFEAT   30
#define NCELLS  (8192 * 7 * 7)        // 401408
#define TPB     256
#define NBLK    (NCELLS / TPB)        // 1568 (exact)
#define NACC    11                    // 8 sums + 3 counts
#define NPAD    1600                  // 25 * 64 for the WMMA reduction

typedef unsigned int v4u  __attribute__((ext_vector_type(4)));
typedef int          v8i_t __attribute__((ext_vector_type(8)));
typedef int          v4i_t __attribute__((ext_vector_type(4)));
typedef float        v2f  __attribute__((ext_vector_type(2)));
typedef float        v8f  __attribute__((ext_vector_type(8)));

#if defined(__gfx1250__) && __has_builtin(__builtin_amdgcn_tensor_load_to_lds) && __has_builtin(__builtin_amdgcn_s_wait_tensorcnt)
#define USE_TDM 1
#else
#define USE_TDM 0
#endif

#if defined(__gfx1250__) && __has_builtin(__builtin_amdgcn_wmma_f32_16x16x4_f32)
#define USE_WMMA 1
#else
#define USE_WMMA 0
#endif

#if USE_TDM
// Flat LDS pointers carry the workgroup-relative byte offset in addr[31:0]
// (ISA 10.2 aperture translation).
__device__ static inline unsigned lds_off_u32(const void* p) {
  return (unsigned)(uintptr_t)p;
}

// One 1-D contiguous TDM load of `nelem` 4-byte elements into LDS.
// D# built per CDNA5 ISA (groups 0..3 bit layout).
__device__ static inline void tdm_load_1d(unsigned ldsOff, const void* gsrc, unsigned nelem) {
  unsigned long long ga = (unsigned long long)(uintptr_t)gsrc;
  v4u g0;
  g0.x = 1u;                                  // count=1 (valid), is_restore=0, gather=0
  g0.y = ldsOff;                              // lds_addr (bytes)
  g0.z = (unsigned)ga;                        // global_addr[31:0]
  g0.w = (unsigned)(ga >> 32) | (2u << 30);   // global_addr[56:32] | type=2 ("image")
  v8i_t g1;
  g1[0] = (int)(2u << 16);                    // wg_mask=0, data_size=2 (4B), no pad/iterate
  g1[1] = (int)((nelem & 0xFFFFu) << 16);     // atomic_bar=0 | tensor_dim0[15:0]
  g1[2] = (int)((nelem >> 16) | (1u << 16));  // tensor_dim0[31:16] | tensor_dim1[15:0]=1
  g1[3] = (int)((nelem & 0xFFFFu) << 16);     // tensor_dim1[31:16]=0 | tile_dim0=nelem
  g1[4] = 1;                                  // tile_dim1=1, tile_dim2=0 (unused)
  g1[5] = (int)nelem;                         // tensor_dim0_stride[31:0]
  g1[6] = 0;                                  // stride0[47:32] | stride1[15:0]
  g1[7] = 0;                                  // stride1[47:16]
#if __has_builtin(__builtin_amdgcn_tensor_load_to_lds_d2)
  // <=2D descriptor form (ISA: VADDR2/VADDR3 = NULL) — exactly our case.
  __builtin_amdgcn_tensor_load_to_lds_d2(g0, g1, 0);
#else
  v4i_t g2; g2[0] = 1; g2[1] = 1; g2[2] = 0; g2[3] = 0;               // dim2=1, dim3=1, stride2=0, tile3=0
  v4i_t g3; g3[0] = 0; g3[1] = (int)(1u << 16); g3[2] = 0; g3[3] = 0; // stride3=0, dim4=1, tile4=0
  v8i_t gz; gz[0]=0; gz[1]=0; gz[2]=0; gz[3]=0; gz[4]=0; gz[5]=0; gz[6]=0; gz[7]=0;
  // clang-23 6-arg form: (uint32x4, int32x8, int32x4, int32x4, int32x8, i32 cpol)
  __builtin_amdgcn_tensor_load_to_lds(g0, g1, g2, g3, gz, 0);
#endif
}
#endif

// ---------------- pass 1: per-block partial sums ----------------
// acc layout: 0 S_loc1, 1 S_loc2, 2 S_obj1, 3 S_obj2, 4 S_cls1, 5 S_cls2,
//             6 S_no4, 7 S_no9, 8 C1, 9 C2, 10 Cno
__global__ void __launch_bounds__(TPB)
yolo_partials(const float* __restrict__ pred, const float* __restrict__ targ,
              const int* __restrict__ coin, float* __restrict__ ws) {
  __shared__ __align__(16) float sp[TPB * DFEAT];   // 30720 B
  __shared__ __align__(16) float st[TPB * DFEAT];   // 30720 B
  __shared__ __align__(16) int   sc[TPB];           // 1024 B
  __shared__ float wred[NACC][TPB / 32];

  const int tid = threadIdx.x;
  const int bid = blockIdx.x;
  const size_t cell0 = (size_t)bid * TPB;

#if USE_TDM
  if (tid < 32) {  // wave 0 issues the three DMA descriptors
    tdm_load_1d(lds_off_u32(sp), pred + cell0 * DFEAT, TPB * DFEAT);
    tdm_load_1d(lds_off_u32(st), targ + cell0 * DFEAT, TPB * DFEAT);
    tdm_load_1d(lds_off_u32(sc), coin + cell0, TPB);
    __builtin_amdgcn_s_wait_tensorcnt(0);
  }
#else
  {
    const float4* gp = (const float4*)(pred + cell0 * DFEAT);
    const float4* gt = (const float4*)(targ + cell0 * DFEAT);
    float4* lp = (float4*)sp;
    float4* lt = (float4*)st;
    for (int i = tid; i < TPB * DFEAT / 4; i += TPB) { lp[i] = gp[i]; lt[i] = gt[i]; }
    const int4* gc = (const int4*)(coin + cell0);
    int4* lc = (int4*)sc;
    if (tid < TPB / 4) lc[tid] = gc[tid];
  }
#endif
  __syncthreads();

  const float* tp = sp + tid * DFEAT;
  const float* tt = st + tid * DFEAT;

  float acc[NACC];
#pragma unroll
  for (int a = 0; a < NACC; ++a) acc[a] = 0.f;

  float loc1 = 0.f, loc2 = 0.f, cls = 0.f;
#pragma unroll
  for (int k = 0; k < 4; ++k)  { float d = tp[k] - tt[k]; loc1 += d * d; }
#pragma unroll
  for (int k = 5; k < 9; ++k)  { float d = tp[k] - tt[k]; loc2 += d * d; }
#pragma unroll
  for (int k = 10; k < 30; ++k){ float d = tp[k] - tt[k]; cls  += d * d; }

  const float p4 = tp[4], p9 = tp[9];
  const bool obj = tt[4] > 0.f;      // target[...,4] == target[...,9] == obj flag
  const bool ch9 = sc[tid] > 0;      // coin>0 -> box1 responsible (t9 zeroed)
  if (obj) {
    if (ch9) { acc[0] = loc1; acc[2] = (p4 - 1.f) * (p4 - 1.f); acc[4] = cls; acc[8] = 1.f; }
    else     { acc[1] = loc2; acc[3] = (p9 - 1.f) * (p9 - 1.f); acc[5] = cls; acc[9] = 1.f; }
  } else {
    acc[6] = p4 * p4; acc[7] = p9 * p9; acc[10] = 1.f;
  }

  const int lane = tid & 31, wv = tid >> 5;
#pragma unroll
  for (int a = 0; a < NACC; ++a) {
    float v = acc[a];
#pragma unroll
    for (int off = 16; off > 0; off >>= 1) v += __shfl_down(v, off, 32);
    if (lane == 0) wred[a][wv] = v;
  }
  __syncthreads();
  if (tid < NACC) {
    float s = 0.f;
#pragma unroll
    for (int w = 0; w < TPB / 32; ++w) s += wred[tid][w];
    ws[tid * NBLK + bid] = s;   // SoA: acc-major
  }
}

// ---------------- pass 2: WMMA reduction tree + final math ----------------
__global__ void __launch_bounds__(32)
yolo_finalize(const float* __restrict__ ws, float* __restrict__ out) {
  const int lane = threadIdx.x;   // single wave, EXEC all ones
  float s[NACC];
#pragma unroll
  for (int a = 0; a < NACC; ++a) s[a] = 0.f;

#if USE_WMMA
  for (int a = 0; a < NACC; ++a) {
    v8f c = {0.f, 0.f, 0.f, 0.f, 0.f, 0.f, 0.f, 0.f};
    v2f ones; ones.x = 1.f; ones.y = 1.f;       // B = all-ones: layout-agnostic
    for (int base = 0; base < NPAD; base += 64) {
      // A is 16x4 f32: lanes 0-15 hold A[M=lane, K=0..1], lanes 16-31 hold A[M=lane-16, K=2..3]
      int e0 = base + 4 * (lane & 15) + 2 * (lane >> 4);
      float x0 = (e0 < NBLK)     ? ws[a * NBLK + e0]     : 0.f;
      float x1 = (e0 + 1 < NBLK) ? ws[a * NBLK + e0 + 1] : 0.f;
      v2f av; av.x = x0; av.y = x1;
      // D[m,n] = sum_k A[m,k] + C[m,n]: 64 new elements folded into 16 row-sums
      c = __builtin_amdgcn_wmma_f32_16x16x4_f32(false, av, false, ones,
                                                (short)0, c, false, false);
    }
    float tot = 0.f;
#pragma unroll
    for (int r = 0; r < 8; ++r) {               // c[r]: lanes0-15 -> M=r, lanes16-31 -> M=r+8
      tot += __shfl(c[r], 0, 32);
      tot += __shfl(c[r], 16, 32);
    }
    s[a] = tot;
  }
#else
  if (lane == 0) {
    for (int a = 0; a < NACC; ++a) {
      double t = 0.0;
      for (int i = 0; i < NBLK; ++i) t += (double)ws[a * NBLK + i];
      s[a] = (float)t;
    }
  }
#endif

  if (lane == 0) {
    const float C1 = fmaxf(s[8], 1.f);
    const float C2 = fmaxf(s[9], 1.f);
    const float Cn = fmaxf(s[10], 1.f);
    const float loc = s[0] / (4.f * C1) + s[1] / (4.f * C2);
    const float obj = s[2] / C1 + s[3] / C2;
    const float cls = s[4] / (20.f * C1) + s[5] / (20.f * C2);
    const float noo = s[6] / Cn + s[7] / Cn;
    const float invB = 1.0f / 8192.0f;
    out[0] = 5.0f * loc * invB;   // L_COORD * loc_loss / batch
    out[1] = cls * invB;          // cls_loss / batch
    out[2] = obj * invB;          // obj_loss / batch
    out[3] = 0.5f * noo * invB;   // L_NOOBJ * noobj_loss / batch
  }
}

// ---------------- entry point ----------------
extern "C" void kernel_launch(void* const* d_in, const int* in_sizes, int n_in,
                              void* d_out, int out_size, void* d_ws, size_t ws_size,
                              hipStream_t stream) {
  const float* pred = (const float*)d_in[0];
  const float* targ = (const float*)d_in[1];
  const int*   coin = (const int*)d_in[2];
  float* out = (float*)d_out;
  float* ws  = (float*)d_ws;   // needs NACC*NBLK*4 = 68,992 bytes -> ~69 KB

  yolo_partials<<<dim3(NBLK), dim3(TPB), 0, stream>>>(pred, targ, coin, ws);
  yolo_finalize<<<dim3(1), dim3(32), 0, stream>>>(ws, out);

  (void)in_sizes; (void)n_in; (void)out_size; (void)ws_size;
}